// SF_DDI_11647951307196
// MI455X (gfx1250) — compile-verified
//
#include <hip/hip_runtime.h>
#include <hip/hip_bf16.h>
#include <math.h>

#define NN   40000
#define NE   150000
#define NLG  450000
#define NG   1500
#define FD   64
#define SNDW 384
#define NIT  10
#define KC   64          // K-chunk staged in LDS
#define LDSS (KC + 2)    // padded LDS row stride (even -> 8B-aligned float2 loads)

typedef __attribute__((ext_vector_type(2))) float v2f;
typedef __attribute__((ext_vector_type(8))) float v8f;

// ---------------------------------------------------------------------------
// WMMA f32 GEMM: C[M,N] = preop(A)[M,K] @ W[K,N] + bias
// preop: mode 0 = identity; 1 = batchnorm; 2 = batchnorm + PReLU
// block = 128 threads (4 waves); grid = (M/64, N/64).
// Wave w computes rows [blockIdx.x*64 + 16w, +16) x cols [blockIdx.y*64, +64)
// using 4 v8f accumulators. A and W staged per 64-K chunk into LDS; W is
// stored transposed (n-major) so B fragments are single aligned ds_load_b64.
// ---------------------------------------------------------------------------
__global__ void gemm_bn_wmma(const float* __restrict__ A, const float* __restrict__ W,
                             const float* __restrict__ bias, float* __restrict__ C,
                             int M, int K, int N,
                             const float* __restrict__ mu, const float* __restrict__ rs,
                             const float* __restrict__ gam, const float* __restrict__ bet,
                             const float* __restrict__ prelu, int mode)
{
    __shared__ float sA[64 * LDSS];   // 64 rows x KC
    __shared__ float sB[64 * LDSS];   // 64 cols x KC (transposed W)

    const int tid   = threadIdx.x;
    const int wave  = tid >> 5;
    const int lane  = tid & 31;
    const int lrow  = lane & 15;
    const int khalf = (lane >> 4) * 2;          // 0 (lanes 0-15) or 2 (lanes 16-31)
    const int m0    = blockIdx.x * 64;
    const int bn0   = blockIdx.y * 64;
    const float pw  = (mode == 2) ? prelu[0] : 0.f;

    v8f acc[4] = {};

    for (int kc = 0; kc < K; kc += KC) {
        // ---- stage A chunk (64 rows x 64 k) with fused pre-op ----
        for (int it = tid; it < 64 * 16; it += 128) {
            int r = it >> 4, q = it & 15;
            const float4 av = *(const float4*)&A[(size_t)(m0 + r) * K + kc + 4 * q];
            float v[4] = {av.x, av.y, av.z, av.w};
            if (mode >= 1) {
                int kb = kc + 4 * q;
#pragma unroll
                for (int c = 0; c < 4; ++c) {
                    float t2 = (v[c] - mu[kb + c]) * rs[kb + c] * gam[kb + c] + bet[kb + c];
                    if (mode == 2 && t2 < 0.f) t2 *= pw;
                    v[c] = t2;
                }
            }
#pragma unroll
            for (int c = 0; c < 4; ++c) sA[r * LDSS + 4 * q + c] = v[c];
        }
        // ---- stage W chunk transposed: sB[n][k] = W[kc+k][bn0+n] ----
        for (int it = tid; it < 64 * 16; it += 128) {
            int k = it >> 4, n4 = it & 15;
            const float4 wv = *(const float4*)&W[(size_t)(kc + k) * N + bn0 + 4 * n4];
            sB[(4 * n4 + 0) * LDSS + k] = wv.x;
            sB[(4 * n4 + 1) * LDSS + k] = wv.y;
            sB[(4 * n4 + 2) * LDSS + k] = wv.z;
            sB[(4 * n4 + 3) * LDSS + k] = wv.w;
        }
        __syncthreads();

        // ---- compute: 16 k-steps, 1 A-frag + 4 B-frags + 4 WMMA each ----
        const int aoff = (wave * 16 + lrow) * LDSS + khalf;
#pragma unroll 4
        for (int k = 0; k < KC; k += 4) {
            v2f a = *(const v2f*)&sA[aoff + k];
#pragma unroll
            for (int j = 0; j < 4; ++j) {
                v2f b = *(const v2f*)&sB[(16 * j + lrow) * LDSS + khalf + k];
                acc[j] = __builtin_amdgcn_wmma_f32_16x16x4_f32(false, a, false, b,
                                                               (short)0, acc[j], false, false);
            }
        }
        __syncthreads();
    }

    // C/D layout: VGPR r -> M = r (lanes 0-15) or M = r+8 (lanes 16-31), N = lane&15
    const int rowBase = m0 + wave * 16 + ((lane >> 4) << 3);
#pragma unroll
    for (int j = 0; j < 4; ++j) {
        int col  = bn0 + 16 * j + lrow;
        float bc = bias ? bias[col] : 0.f;
#pragma unroll
        for (int r = 0; r < 8; ++r)
            C[(size_t)(rowBase + r) * N + col] = acc[j][r] + bc;
    }
}

// ---------------------------------------------------------------------------
// edge_attr[e,f] = (U[src,f] + V[dst,f] + sum_d ear[e,d]*We[d,f]) / 3
// ---------------------------------------------------------------------------
__global__ void edge_init_kernel(const float* __restrict__ U, const float* __restrict__ V,
                                 const float* __restrict__ ear, const float* __restrict__ We,
                                 const int* __restrict__ ei, float* __restrict__ ea)
{
    int gid = blockIdx.x * blockDim.x + threadIdx.x;
    if (gid >= NE * FD) return;
    int e = gid >> 6, f = gid & 63;
    int s = ei[e], d = ei[NE + e];
    float uv = 0.f;
#pragma unroll
    for (int k = 0; k < 6; ++k) uv += ear[e * 6 + k] * We[k * FD + f];
    ea[gid] = (U[(size_t)s * FD + f] + V[(size_t)d * FD + f] + uv) * (1.f / 3.f);
}

// scatter over line graph: 2 features per thread (one index load pair per 2 adds)
__global__ void scatter_lg_kernel(const int* __restrict__ lge, const float* __restrict__ out,
                                  float* __restrict__ agg)
{
    int gid = blockIdx.x * blockDim.x + threadIdx.x;
    if (gid >= NLG * 32) return;
    int j = gid >> 5, f = gid & 31;
    int src = lge[j], dst = lge[NLG + j];
    size_t sb = (size_t)src * FD + f, db = (size_t)dst * FD + f;
    atomicAdd(&agg[db],      out[sb]);
    atomicAdd(&agg[db + 32], out[sb + 32]);
}

__global__ void scatter_scalar_kernel(const int* __restrict__ lge, const float* __restrict__ r,
                                      float* __restrict__ m)
{
    int j = blockIdx.x * blockDim.x + threadIdx.x;
    if (j >= NLG) return;
    atomicAdd(&m[lge[NLG + j]], r[lge[j]]);
}

// out = ea + agg ; r = out . Wrel ; rt = out . Wroot  (one warp per edge)
__global__ void finish_out_kernel(const float* __restrict__ ea, const float* __restrict__ agg,
                                  const float* __restrict__ Wrel, const float* __restrict__ Wroot,
                                  float* __restrict__ out, float* __restrict__ r,
                                  float* __restrict__ rt)
{
    int warp = (blockIdx.x * blockDim.x + threadIdx.x) >> 5;
    int lane = threadIdx.x & 31;
    if (warp >= NE) return;
    size_t base = (size_t)warp * FD;
    float o0 = ea[base + lane]      + agg[base + lane];
    float o1 = ea[base + lane + 32] + agg[base + lane + 32];
    out[base + lane]      = o0;
    out[base + lane + 32] = o1;
    float rr = o0 * Wrel[lane]  + o1 * Wrel[lane + 32];
    float ro = o0 * Wroot[lane] + o1 * Wroot[lane + 32];
    for (int off = 16; off > 0; off >>= 1) {
        rr += __shfl_xor(rr, off);
        ro += __shfl_xor(ro, off);
    }
    if (lane == 0) { r[warp] = rr; rt[warp] = ro; }
}

__device__ inline void atomicMaxF(float* addr, float val)
{
    int old = __float_as_int(*addr);
    while (__int_as_float(old) < val) {
        int prev = atomicCAS((int*)addr, old, __float_as_int(val));
        if (prev == old) break;
        old = prev;
    }
}

__global__ void init_graphs_kernel(float* gmax, float* gsum)
{
    int g = blockIdx.x * blockDim.x + threadIdx.x;
    if (g >= NG) return;
    gmax[g] = -INFINITY;
    gsum[g] = 0.f;
}

__global__ void xconv_max_kernel(const float* __restrict__ m, const float* __restrict__ rt,
                                 const float* __restrict__ brel, const int* __restrict__ batch,
                                 float* __restrict__ xc, float* __restrict__ gmax)
{
    int e = blockIdx.x * blockDim.x + threadIdx.x;
    if (e >= NE) return;
    float v = m[e] + brel[0] + rt[e];
    xc[e] = v;
    atomicMaxF(&gmax[batch[e]], v);
}

__global__ void expsum_kernel(const float* __restrict__ xc, const float* __restrict__ gmax,
                              const int* __restrict__ batch, float* __restrict__ ev,
                              float* __restrict__ gsum)
{
    int e = blockIdx.x * blockDim.x + threadIdx.x;
    if (e >= NE) return;
    int b = batch[e];
    float v = expf(xc[e] - gmax[b]);
    ev[e] = v;
    atomicAdd(&gsum[b], v);
}

__global__ void gout_accum_kernel(const float* __restrict__ out, const float* __restrict__ ev,
                                  const float* __restrict__ gsum, const int* __restrict__ batch,
                                  float* __restrict__ gout)
{
    int gid = blockIdx.x * blockDim.x + threadIdx.x;
    if (gid >= NE * FD) return;
    int e = gid >> 6, f = gid & 63;
    int b = batch[e];
    float w = ev[e] / (gsum[b] + 1e-16f);
    atomicAdd(&gout[b * FD + f], out[gid] * w);
}

// sc[g,t] = sum_f tanh((gout@Wg + bg))[g,f] * a[f,t] + a_bias[t]  (one warp per graph)
__global__ void gscore_kernel(const float* __restrict__ gout, const float* __restrict__ Wg,
                              const float* __restrict__ bg, const float* __restrict__ a,
                              const float* __restrict__ a_bias, float* __restrict__ sc, int t)
{
    int warp = (blockIdx.x * blockDim.x + threadIdx.x) >> 5;
    int lane = threadIdx.x & 31;
    if (warp >= NG) return;
    const float* gr = gout + (size_t)warp * FD;
    float acc = 0.f;
#pragma unroll
    for (int h = 0; h < 2; ++h) {
        int f = lane + h * 32;
        float s = bg[f];
        for (int k = 0; k < FD; ++k) s += gr[k] * Wg[k * FD + f];
        acc += tanhf(s) * a[f * NIT + t];
    }
    for (int off = 16; off > 0; off >>= 1) acc += __shfl_xor(acc, off);
    if (lane == 0) sc[warp * NIT + t] = acc + a_bias[t];
}

__global__ void softmax_t_kernel(const float* __restrict__ sc, float* __restrict__ alpha)
{
    int g = blockIdx.x * blockDim.x + threadIdx.x;
    if (g >= NG) return;
    float m = -INFINITY;
    for (int t = 0; t < NIT; ++t) m = fmaxf(m, sc[g * NIT + t]);
    float e[NIT], s = 0.f;
    for (int t = 0; t < NIT; ++t) { e[t] = expf(sc[g * NIT + t] - m); s += e[t]; }
    for (int t = 0; t < NIT; ++t) alpha[g * NIT + t] = e[t] / s;
}

// pass 2: out = ea + agg ; acc += out * alpha[batch, t]
__global__ void accum_kernel(const float* __restrict__ ea, const float* __restrict__ agg,
                             const float* __restrict__ alpha, const int* __restrict__ batch,
                             int t, float* __restrict__ out, float* __restrict__ acc)
{
    int gid = blockIdx.x * blockDim.x + threadIdx.x;
    if (gid >= NE * FD) return;
    int e = gid >> 6;
    float o = ea[gid] + agg[gid];
    out[gid] = o;
    acc[gid] += o * alpha[batch[e] * NIT + t];
}

__global__ void node_scatter_kernel(const int* __restrict__ ei, const float* __restrict__ acc,
                                    float* __restrict__ xn)
{
    int gid = blockIdx.x * blockDim.x + threadIdx.x;
    if (gid >= NE * FD) return;
    int e = gid >> 6, f = gid & 63;
    int d = ei[NE + e];
    atomicAdd(&xn[(size_t)d * FD + f], acc[gid]);
}

__global__ void avg_kernel(const float* __restrict__ a, float* __restrict__ b, int n)
{
    int i = blockIdx.x * blockDim.x + threadIdx.x;
    if (i >= n) return;
    b[i] = (a[i] + b[i]) * 0.5f;
}

// column sums for batchnorm stats
__global__ void colstat_kernel(const float* __restrict__ X, int M, int C,
                               float* __restrict__ sum, float* __restrict__ sq)
{
    int c = blockIdx.x * 64 + threadIdx.x;
    float s = 0.f, s2 = 0.f;
    for (int r = blockIdx.y; r < M; r += gridDim.y) {
        float v = X[(size_t)r * C + c];
        s += v; s2 += v * v;
    }
    atomicAdd(&sum[c], s);
    atomicAdd(&sq[c], s2);
}

__global__ void bnfinal_kernel(const float* __restrict__ sum, const float* __restrict__ sq,
                               int M, int C, float* __restrict__ mu, float* __restrict__ rs)
{
    int c = blockIdx.x * blockDim.x + threadIdx.x;
    if (c >= C) return;
    float m = sum[c] / (float)M;
    float v = sq[c] / (float)M - m * m;
    mu[c] = m;
    rs[c] = rsqrtf(v + 1e-5f);
}

// ---------------------------------------------------------------------------
extern "C" void kernel_launch(void* const* d_in, const int* in_sizes, int n_in,
                              void* d_out, int out_size, void* d_ws, size_t ws_size,
                              hipStream_t stream)
{
    (void)in_sizes; (void)n_in; (void)out_size; (void)ws_size;
    // inputs (setup_inputs dict order, params flattened in insertion order)
    const float* x    = (const float*)d_in[0];
    const float* ear  = (const float*)d_in[1];
    const float* Wu   = (const float*)d_in[2];
    const float* Wv   = (const float*)d_in[3];
    const float* We   = (const float*)d_in[4];
    const float* Wrel = (const float*)d_in[5];
    const float* brel = (const float*)d_in[6];
    const float* Wroot= (const float*)d_in[7];
    const float* Wg   = (const float*)d_in[8];
    const float* bg   = (const float*)d_in[9];
    const float* avec = (const float*)d_in[10];
    const float* abia = (const float*)d_in[11];
    const float* W1 = (const float*)d_in[12]; const float* b1 = (const float*)d_in[13];
    const float* W2 = (const float*)d_in[14]; const float* b2 = (const float*)d_in[15];
    const float* W3 = (const float*)d_in[16]; const float* b3 = (const float*)d_in[17];
    const float* W4 = (const float*)d_in[18]; const float* b4 = (const float*)d_in[19];
    const float* W5 = (const float*)d_in[20]; const float* b5 = (const float*)d_in[21];
    const float* bn1g = (const float*)d_in[22]; const float* bn1b = (const float*)d_in[23];
    const float* bn2g = (const float*)d_in[24]; const float* bn2b = (const float*)d_in[25];
    const float* bn3g = (const float*)d_in[26]; const float* bn3b = (const float*)d_in[27];
    const float* bn4g = (const float*)d_in[28]; const float* bn4b = (const float*)d_in[29];
    const float* bn5g = (const float*)d_in[30]; const float* bn5b = (const float*)d_in[31];
    const float* pr2 = (const float*)d_in[32]; const float* pr3 = (const float*)d_in[33];
    const float* pr4 = (const float*)d_in[34]; const float* pr5 = (const float*)d_in[35];
    const int* ei    = (const int*)d_in[36];
    const int* lge   = (const int*)d_in[37];
    const int* batch = (const int*)d_in[38];

    // workspace layout (floats)
    float* ws = (float*)d_ws;
    size_t off = 0;
    float* EA  = ws + off; off += (size_t)NE * FD;
    float* OUT = ws + off; off += (size_t)NE * FD;
    float* AGG = ws + off; off += (size_t)NE * FD;
    float* ACC = ws + off; off += (size_t)NE * FD;
    float* HA  = ws + off; off += (size_t)NN * SNDW;   // also holds U | V early
    float* HB  = ws + off; off += (size_t)NN * SNDW;
    float* HC  = ws + off; off += (size_t)NN * SNDW;   // also holds XN early
    float* RE  = ws + off; off += NE;
    float* RT  = ws + off; off += NE;
    float* MM  = ws + off; off += NE;
    float* XC  = ws + off; off += NE;
    float* EV  = ws + off; off += NE;
    float* GMAX= ws + off; off += NG;
    float* GSUM= ws + off; off += NG;
    float* GOUT= ws + off; off += (size_t)NG * FD;
    float* SC  = ws + off; off += (size_t)NG * NIT;
    float* ALP = ws + off; off += (size_t)NG * NIT;
    float* BSUM= ws + off; off += SNDW;
    float* BSQ = ws + off; off += SNDW;
    float* BMU = ws + off; off += SNDW;
    float* BRS = ws + off; off += SNDW;

    float* U  = HA;
    float* V  = HA + (size_t)NN * FD;
    float* XN = HC;

    auto gemm = [&](const float* A, const float* W, const float* bias, float* C,
                    int M, int K, int N, int mode,
                    const float* g, const float* b, const float* pr) {
        dim3 grid(M / 64, N / 64);
        gemm_bn_wmma<<<grid, 128, 0, stream>>>(A, W, bias, C, M, K, N,
                                               BMU, BRS, g, b, pr, mode);
    };
    auto stats = [&](const float* X, int M, int C) {
        hipMemsetAsync(BSUM, 0, 2 * SNDW * sizeof(float), stream);  // BSUM + BSQ
        dim3 g(C / 64, 64);
        colstat_kernel<<<g, 64, 0, stream>>>(X, M, C, BSUM, BSQ);
        bnfinal_kernel<<<(C + 63) / 64, 64, 0, stream>>>(BSUM, BSQ, M, C, BMU, BRS);
    };

    const int TB = 256;
    const int nEF  = (NE * FD + TB - 1) / TB;
    const int nLGH = (NLG * 32 + TB - 1) / TB;    // 2 feats per thread
    const int nLG  = (NLG + TB - 1) / TB;
    const int nE   = (NE + TB - 1) / TB;
    const int nEW  = (NE * 32 + TB - 1) / TB;     // warp-per-edge
    const int nGW  = (NG * 32 + TB - 1) / TB;     // warp-per-graph
    const size_t edgeB = (size_t)NE * FD * sizeof(float);

    // ---- projections + edge init ----
    gemm(x, Wu, nullptr, U, NN, FD, FD, 0, nullptr, nullptr, nullptr);
    gemm(x, Wv, nullptr, V, NN, FD, FD, 0, nullptr, nullptr, nullptr);
    edge_init_kernel<<<nEF, TB, 0, stream>>>(U, V, ear, We, ei, EA);

    // ---- pass 1: message passing + per-iteration attention logits ----
    hipMemcpyAsync(OUT, EA, edgeB, hipMemcpyDeviceToDevice, stream);
    for (int t = 0; t < NIT; ++t) {
        hipMemsetAsync(AGG, 0, edgeB, stream);
        scatter_lg_kernel<<<nLGH, TB, 0, stream>>>(lge, OUT, AGG);
        finish_out_kernel<<<nEW, TB, 0, stream>>>(EA, AGG, Wrel, Wroot, OUT, RE, RT);
        hipMemsetAsync(MM, 0, NE * sizeof(float), stream);
        scatter_scalar_kernel<<<nLG, TB, 0, stream>>>(lge, RE, MM);
        init_graphs_kernel<<<(NG + TB - 1) / TB, TB, 0, stream>>>(GMAX, GSUM);
        xconv_max_kernel<<<nE, TB, 0, stream>>>(MM, RT, brel, batch, XC, GMAX);
        expsum_kernel<<<nE, TB, 0, stream>>>(XC, GMAX, batch, EV, GSUM);
        hipMemsetAsync(GOUT, 0, (size_t)NG * FD * sizeof(float), stream);
        gout_accum_kernel<<<nEF, TB, 0, stream>>>(OUT, EV, GSUM, batch, GOUT);
        gscore_kernel<<<nGW, TB, 0, stream>>>(GOUT, Wg, bg, avec, abia, SC, t);
    }
    softmax_t_kernel<<<(NG + TB - 1) / TB, TB, 0, stream>>>(SC, ALP);

    // ---- pass 2: replay recurrence, accumulate alpha-weighted sum ----
    hipMemcpyAsync(OUT, EA, edgeB, hipMemcpyDeviceToDevice, stream);
    hipMemsetAsync(ACC, 0, edgeB, stream);
    for (int t = 0; t < NIT; ++t) {
        hipMemsetAsync(AGG, 0, edgeB, stream);
        scatter_lg_kernel<<<nLGH, TB, 0, stream>>>(lge, OUT, AGG);
        accum_kernel<<<nEF, TB, 0, stream>>>(EA, AGG, ALP, batch, t, OUT, ACC);
    }

    // ---- node update: xn = x + scatter_add(acc by dst) ----
    hipMemcpyAsync(XN, x, (size_t)NN * FD * sizeof(float), hipMemcpyDeviceToDevice, stream);
    node_scatter_kernel<<<nEF, TB, 0, stream>>>(ei, ACC, XN);

    // ---- LinearBlock (all GEMMs via WMMA with fused BN/PReLU pre-op) ----
    stats(XN, NN, FD);
    gemm(XN, W1, b1, HA, NN, FD, SNDW, 1, bn1g, bn1b, nullptr);          // h1
    stats(HA, NN, SNDW);
    gemm(HA, W2, b2, HB, NN, SNDW, SNDW, 2, bn2g, bn2b, pr2);            // h2
    stats(HB, NN, SNDW);
    gemm(HB, W3, b3, HC, NN, SNDW, SNDW, 2, bn3g, bn3b, pr3);            // h3 (XN dead)
    avg_kernel<<<(NN * SNDW + TB - 1) / TB, TB, 0, stream>>>(HC, HA, NN * SNDW); // x2 in HA
    stats(HA, NN, SNDW);
    gemm(HA, W4, b4, HB, NN, SNDW, SNDW, 2, bn4g, bn4b, pr4);            // h4
    avg_kernel<<<(NN * SNDW + TB - 1) / TB, TB, 0, stream>>>(HB, HA, NN * SNDW); // x3 in HA
    stats(HA, NN, SNDW);
    gemm(HA, W5, b5, (float*)d_out, NN, SNDW, FD, 2, bn5g, bn5b, pr5);   // final [NN,FD]
}